// Fcaf3DNeckWithHead_20306605375872
// MI455X (gfx1250) — compile-verified
//
#include <hip/hip_runtime.h>
#include <stdint.h>

typedef float v2f __attribute__((ext_vector_type(2)));
typedef float v8f __attribute__((ext_vector_type(8)));

#define K_SEL     50000u   // PTS_THRESHOLD
#define NBATCH    8
#define CDIM      128
#define NCOL      32       // padded GEMM output columns (27 used)
#define LDK       130      // padded K stride for weights in LDS (bank-conflict avoidance)

// Monotonic float->uint key (ascending order preserved)
__device__ __forceinline__ unsigned fkey(float s) {
  unsigned u = __float_as_uint(s);
  return u ^ (((int)u < 0) ? 0xFFFFFFFFu : 0x80000000u);
}

// ---------------- weight packing: W_t[c][k] = [W_center | W_reg | W_cls](k, c) ----------------
__global__ void fcaf_pack_wt(const float* __restrict__ Wc, const float* __restrict__ Wr,
                             const float* __restrict__ Wl, float* __restrict__ wt) {
  int t = blockIdx.x * blockDim.x + threadIdx.x;
  if (t >= NCOL * CDIM) return;
  int c = t >> 7, k = t & 127;
  float v = 0.0f;
  if (c == 0)       v = Wc[k];
  else if (c <= 8)  v = Wr[k * 8 + (c - 1)];
  else if (c <= 26) v = Wl[k * 18 + (c - 9)];
  wt[c * CDIM + k] = v;
}

// ---------------- radix select (exact per-batch k-th largest score key) ----------------
__global__ void fcaf_sel_init(unsigned* __restrict__ hist, unsigned* __restrict__ prefix,
                              unsigned* __restrict__ kremain) {
  int b = blockIdx.x, t = threadIdx.x;
  hist[b * 256 + t] = 0u;
  if (t == 0) { prefix[b] = 0u; kremain[b] = K_SEL; }
}

__global__ void fcaf_sel_hist(const float* __restrict__ scores, const int* __restrict__ bidx,
                              unsigned* __restrict__ hist, const unsigned* __restrict__ prefix,
                              int n, int shift) {
  int i = blockIdx.x * blockDim.x + threadIdx.x;
  if (i >= n) return;
  unsigned key = fkey(scores[i]);
  int b = bidx[i];
  bool ok = (shift == 24) || ((key >> (shift + 8)) == prefix[b]);
  if (ok) atomicAdd(&hist[b * 256 + ((key >> shift) & 255u)], 1u);
}

__global__ void fcaf_sel_pick(unsigned* __restrict__ hist, unsigned* __restrict__ prefix,
                              unsigned* __restrict__ kremain, unsigned* __restrict__ thresh,
                              int shift) {
  int b = blockIdx.x;
  if (threadIdx.x == 0) {
    unsigned kr = kremain[b], cum = 0u;
    int d = 0;
    for (int dd = 255; dd >= 0; --dd) {
      unsigned h = hist[b * 256 + dd];
      if (cum + h >= kr) { d = dd; kr = kr - cum; break; }
      cum += h;
    }
    kremain[b] = kr;
    unsigned p = (shift == 24) ? (unsigned)d : ((prefix[b] << 8) | (unsigned)d);
    prefix[b] = p;
    if (shift == 0) thresh[b] = p;
  }
  __syncthreads();
  hist[b * 256 + threadIdx.x] = 0u;   // ready for next pass
}

// ---------------- fused WMMA GEMM + bbox decode + prune mask ----------------
__launch_bounds__(256)
__global__ void fcaf_head_gemm(const float* __restrict__ feats,
                               const float* __restrict__ points,
                               const float* __restrict__ scores,
                               const int*   __restrict__ bidx,
                               const float* __restrict__ wt,      // [32][128]
                               const float* __restrict__ b_cls,   // [18]
                               const float* __restrict__ scale,   // [1]
                               const unsigned* __restrict__ thresh,
                               float* __restrict__ out, int n) {
  __shared__ float sW[NCOL * LDK];          // padded transposed weights
  __shared__ float sO[8 * 16 * NCOL];       // per-wave 16x32 result tiles

  for (int t = threadIdx.x; t < NCOL * CDIM; t += 256) {
    int c = t >> 7, k = t & 127;
    sW[c * LDK + k] = wt[t];
  }
  __syncthreads();

  const int lane = threadIdx.x & 31;
  const int wave = threadIdx.x >> 5;
  const int m  = lane & 15;     // row within tile / column within tile
  const int kh = lane >> 4;     // K-half select per ISA 16x4 f32 A/B layout

  long row  = (long)blockIdx.x * 128 + wave * 16 + m;
  long rowc = (row < n) ? row : (long)(n - 1);   // clamp: EXEC must stay all-ones for WMMA

  const float* arow = feats + rowc * CDIM + 2 * kh;
  const float* bw0  = &sW[m * LDK + 2 * kh];           // cols 0..15
  const float* bw1  = &sW[(16 + m) * LDK + 2 * kh];    // cols 16..31

  v8f c0 = {}; v8f c1 = {};
  #pragma unroll 4
  for (int k0 = 0; k0 < CDIM; k0 += 4) {
    v2f a  = *(const v2f*)(arow + k0);   // A frag: lane holds rows m, K = 2*kh + {0,1}
    v2f b0 = *(const v2f*)(bw0 + k0);    // B frag: lane holds col m, K = 2*kh + {0,1}
    v2f b1 = *(const v2f*)(bw1 + k0);
    c0 = __builtin_amdgcn_wmma_f32_16x16x4_f32(false, a, false, b0, (short)0, c0, false, false);
    c1 = __builtin_amdgcn_wmma_f32_16x16x4_f32(false, a, false, b1, (short)0, c1, false, false);
  }

  // Spill C tiles to wave-private LDS; C layout: VGPR v holds row v + 8*kh, lane&15 = col.
  float* o = &sO[wave * 16 * NCOL];
  #pragma unroll
  for (int v = 0; v < 8; ++v) {
    int r = v + 8 * kh;
    o[r * NCOL + m]      = c0[v];
    o[r * NCOL + 16 + m] = c1[v];
  }
  asm volatile("s_wait_dscnt 0" ::: "memory");   // LDS in-order within wave; no WG barrier needed

  long rrow = (long)blockIdx.x * 128 + wave * 16 + m;
  if (rrow < n) {
    float g[28];
    const float* grow = &o[m * NCOL];
    #pragma unroll
    for (int j = 0; j < 7; ++j) {
      float4 v4 = *(const float4*)(grow + 4 * j);
      g[4*j] = v4.x; g[4*j+1] = v4.y; g[4*j+2] = v4.z; g[4*j+3] = v4.w;
    }
    unsigned key = fkey(scores[rrow]);
    int b = bidx[rrow];
    float keep = (key >= thresh[b]) ? 1.0f : 0.0f;
    float* orow = out + rrow * 26;

    if (kh == 0) {
      // centerness + bbox decode + first 9 cls columns
      float s = scale[0];
      float rd[6];
      #pragma unroll
      for (int j = 0; j < 6; ++j) rd[j] = __expf(s * g[1 + j]);
      float bp6 = g[7], bp7 = g[8];
      float px = points[rrow * 3 + 0] * 0.01f;
      float py = points[rrow * 3 + 1] * 0.01f;
      float pz = points[rrow * 3 + 2] * 0.01f;
      float xc = px + 0.5f * (rd[1] - rd[0]);
      float yc = py + 0.5f * (rd[3] - rd[2]);
      float zc = pz + 0.5f * (rd[5] - rd[4]);
      float scl = rd[0] + rd[1] + rd[2] + rd[3];
      float q = __expf(sqrtf(bp6 * bp6 + bp7 * bp7));
      float w = scl / (1.0f + q);
      float alpha = 0.5f * atan2f(bp6, bp7);
      orow[0] = g[0] * keep;
      orow[1] = xc * keep;
      orow[2] = yc * keep;
      orow[3] = zc * keep;
      orow[4] = w * keep;
      orow[5] = (w * q) * keep;
      orow[6] = (rd[5] + rd[4]) * keep;
      orow[7] = alpha * keep;
      #pragma unroll
      for (int j = 0; j < 9; ++j) orow[8 + j] = (g[9 + j] + b_cls[j]) * keep;
    } else {
      // remaining 9 cls columns
      #pragma unroll
      for (int j = 9; j < 18; ++j) orow[8 + j] = (g[9 + j] + b_cls[j]) * keep;
    }
  }
}

extern "C" void kernel_launch(void* const* d_in, const int* in_sizes, int n_in,
                              void* d_out, int out_size, void* d_ws, size_t ws_size,
                              hipStream_t stream) {
  const float* feats    = (const float*)d_in[0];
  const float* points   = (const float*)d_in[1];
  const float* scores   = (const float*)d_in[2];
  const int*   batchidx = (const int*)  d_in[3];
  const float* W_center = (const float*)d_in[4];
  const float* W_reg    = (const float*)d_in[5];
  const float* W_cls    = (const float*)d_in[6];
  const float* b_cls    = (const float*)d_in[7];
  const float* scale    = (const float*)d_in[8];
  float* out = (float*)d_out;
  const int N = in_sizes[2];   // scores is [N]

  // workspace layout
  float*    wt      = (float*)d_ws;                                   // 32*128 f32 = 16 KB
  unsigned* hist    = (unsigned*)((char*)d_ws + NCOL * CDIM * 4);     // 8*256 u32 = 8 KB
  unsigned* prefix  = hist + NBATCH * 256;                            // 8 u32
  unsigned* kremain = prefix + NBATCH;                                // 8 u32
  unsigned* thresh  = kremain + NBATCH;                               // 8 u32

  // 1) pack transposed weights
  fcaf_pack_wt<<<(NCOL * CDIM + 255) / 256, 256, 0, stream>>>(W_center, W_reg, W_cls, wt);

  // 2) exact per-batch radix select of the k-th largest score key
  fcaf_sel_init<<<NBATCH, 256, 0, stream>>>(hist, prefix, kremain);
  const int shifts[4] = {24, 16, 8, 0};
  for (int p = 0; p < 4; ++p) {
    fcaf_sel_hist<<<(N + 255) / 256, 256, 0, stream>>>(scores, batchidx, hist, prefix, N, shifts[p]);
    fcaf_sel_pick<<<NBATCH, 256, 0, stream>>>(hist, prefix, kremain, thresh, shifts[p]);
  }

  // 3) fused WMMA head: 128 rows per 256-thread block (8 waves x 16 rows)
  fcaf_head_gemm<<<(N + 127) / 128, 256, 0, stream>>>(
      feats, points, scores, batchidx, wt, b_cls, scale, thresh, out, N);
}